// GHMC_Loss_12403865550911
// MI455X (gfx1250) — compile-verified
//
#include <hip/hip_runtime.h>
#include <hip/hip_bf16.h>

#define BINS 10
#define NCOL 1000
#define ROWS_PER_BLK 8
#define THREADS 256
#define EPSF 1e-8f

// ---- gfx1250 async global->LDS path ----
// ROCm 7.2 clang declares the builtin with generic int4* pointers:
//   error msg showed param type: __attribute__((__vector_size__(4*sizeof(int)))) int *
typedef int v4i __attribute__((vector_size(16)));

#if defined(__has_builtin)
#if __has_builtin(__builtin_amdgcn_global_load_async_to_lds_b128)
#define HAVE_ASYNC_B128 1
#endif
#if __has_builtin(__builtin_amdgcn_s_wait_asynccnt)
#define HAVE_WAIT_ASYNC 1
#endif
#endif

__device__ __forceinline__ void async_copy_b128(const float* gsrc, float* ldst) {
#ifdef HAVE_ASYNC_B128
    __builtin_amdgcn_global_load_async_to_lds_b128((v4i*)gsrc, (v4i*)ldst, 0, 0);
#else
    unsigned lds_off = (unsigned)(size_t)ldst;   // low 32 bits of generic LDS ptr = LDS byte addr
    asm volatile("global_load_async_to_lds_b128 %0, %1, off"
                 :: "v"(lds_off), "v"(gsrc) : "memory");
#endif
}

__device__ __forceinline__ void wait_async_zero() {
#ifdef HAVE_WAIT_ASYNC
    __builtin_amdgcn_s_wait_asynccnt(0);
#else
    asm volatile("s_wait_asynccnt 0" ::: "memory");
#endif
}

// ---- kernel 1: zero the workspace (10 counts + 10 loss sums) ----
__global__ void ghmc_init(unsigned* __restrict__ cnt, float* __restrict__ lsum,
                          float* __restrict__ out) {
    int t = threadIdx.x;
    if (t < BINS) { cnt[t] = 0u; lsum[t] = 0.0f; }
    if (t == 0)   { out[0] = 0.0f; }
}

// ---- kernel 2: stream rows, per-row softmax-denominator + bin + loss ----
__global__ __launch_bounds__(THREADS) void ghmc_main(
    const float* __restrict__ pred, const int* __restrict__ target,
    unsigned* __restrict__ g_cnt, float* __restrict__ g_lsum) {

    __shared__ float    tile[ROWS_PER_BLK * NCOL];   // 32000 B, rows contiguous
    __shared__ unsigned s_cnt[BINS];
    __shared__ float    s_lsum[BINS];

    const int       tid  = threadIdx.x;
    const long long row0 = (long long)blockIdx.x * ROWS_PER_BLK;
    const float*    gsrc = pred + row0 * NCOL;       // 8 rows are contiguous in memory

    if (tid < BINS) { s_cnt[tid] = 0u; s_lsum[tid] = 0.0f; }

    // Async-stage 8000 contiguous floats = 2000 x 16B transfers (ASYNCcnt path).
    const int NV4 = (ROWS_PER_BLK * NCOL) / 4;       // 2000
    for (int i = tid; i < NV4; i += THREADS) {
        async_copy_b128(gsrc + i * 4, &tile[i * 4]);
    }
    wait_async_zero();       // this wave's async transfers complete
    __syncthreads();         // all waves' transfers visible in LDS

    const int wave  = tid >> 5;
    const int lane  = tid & 31;
    const float* lrow = &tile[wave * NCOL];          // one wave32 per row

    // sum(exp(x)) over the row; float4 LDS reads, no bank conflicts
    float s = 0.0f;
    const float4* lrow4 = (const float4*)lrow;
    for (int j = lane; j < NCOL / 4; j += 32) {
        float4 v = lrow4[j];
        s += __expf(v.x) + __expf(v.y) + __expf(v.z) + __expf(v.w);
    }
    #pragma unroll
    for (int m = 16; m > 0; m >>= 1) s += __shfl_xor(s, m, 32);

    if (lane == 0) {
        const long long row = row0 + wave;
        const int   t    = target[row];
        const float pt   = lrow[t];                  // pred[row, t] from LDS
        const float prob = __expf(pt) / s;           // == softmax(pred)[row, t]
        const float g    = 1.0f - prob;
        int b = (int)floorf(g * (float)BINS);
        b = min(max(b, 0), BINS - 1);
        const float loss = -pt + __logf(s + EPSF);   // log(sum(exp)+eps)
        atomicAdd(&s_cnt[b], 1u);
        atomicAdd(&s_lsum[b], loss);
    }
    __syncthreads();

    // hierarchical flush: <=10 global atomic pairs per block
    if (tid < BINS && s_cnt[tid] > 0u) {
        atomicAdd(&g_cnt[tid],  s_cnt[tid]);
        atomicAdd(&g_lsum[tid], s_lsum[tid]);
    }
}

// ---- kernel 3: combine 10 bins -> scalar ----
__global__ void ghmc_final(const unsigned* __restrict__ cnt,
                           const float* __restrict__ lsum,
                           float* __restrict__ out) {
    if (threadIdx.x == 0) {
        int ne = 0;
        #pragma unroll
        for (int b = 0; b < BINS; ++b) ne += (cnt[b] > 0u) ? 1 : 0;
        const float inv_ne = 1.0f / (float)max(ne, 1);
        float r = 0.0f;
        #pragma unroll
        for (int b = 0; b < BINS; ++b) {
            if (cnt[b] > 0u) r += lsum[b] / (float)cnt[b];
        }
        out[0] = r * inv_ne;
    }
}

extern "C" void kernel_launch(void* const* d_in, const int* in_sizes, int n_in,
                              void* d_out, int out_size, void* d_ws, size_t ws_size,
                              hipStream_t stream) {
    const float* pred   = (const float*)d_in[0];
    const int*   target = (const int*)d_in[1];
    float*       out    = (float*)d_out;

    unsigned* cnt  = (unsigned*)d_ws;
    float*    lsum = (float*)((char*)d_ws + BINS * sizeof(unsigned));

    const int B      = in_sizes[1];              // 65536 rows
    const int nblock = B / ROWS_PER_BLK;         // 8192

    ghmc_init <<<1, 32, 0, stream>>>(cnt, lsum, out);
    ghmc_main <<<nblock, THREADS, 0, stream>>>(pred, target, cnt, lsum);
    ghmc_final<<<1, 32, 0, stream>>>(cnt, lsum, out);
}